// NonMaximaSuppression2d_37434934952181
// MI455X (gfx1250) — compile-verified
//
#include <hip/hip_runtime.h>
#include <cstdint>
#include <cstddef>

// NMS 3x3 (replicate pad, center-excluded max that implicitly includes 0)
// x: (8, 64, 256, 256) fp32 -> out same shape.
//
// Memory-bound kernel: floor ~268 MB / 23.3 TB/s ~= 11.5 us on MI455X.
// Strategy: TDM (tensor_load_to_lds) stages a 256-wide x (ROWS+halo)-tall fp32
// tile into LDS per workgroup; all 9-point stencil reads hit LDS with
// conflict-free lane-stride-1 access; output written with coalesced b32 stores.

#define IMG_H 256
#define IMG_W 256
#define ROWS  16            // output rows per block
#define NTHREADS 512        // 16 waves, one wave per output row

typedef unsigned int v4u __attribute__((ext_vector_type(4)));
typedef int          v8i __attribute__((ext_vector_type(8)));
typedef int          v4i __attribute__((ext_vector_type(4)));

__global__ __launch_bounds__(NTHREADS) void nms3x3_kernel(const float* __restrict__ x,
                                                          float* __restrict__ out) {
    __shared__ float tile[(ROWS + 2) * IMG_W];   // 18 KB

    const int bid    = blockIdx.x;
    const int ty     = bid & (IMG_H / ROWS - 1);   // 16 row-tiles per plane
    const int plane  = bid >> 4;                   // n*c plane index
    const int y0     = ty * ROWS;
    const int ystart = (y0 == 0) ? 0 : (y0 - 1);
    const int yend   = (y0 + ROWS >= IMG_H) ? (IMG_H - 1) : (y0 + ROWS);
    const int nrows  = yend - ystart + 1;          // 17 (edge tiles) or 18

    const uint64_t ga =
        (uint64_t)(uintptr_t)(x + (size_t)plane * IMG_H * IMG_W + (size_t)ystart * IMG_W);

    // ---- TDM: DMA the (nrows x 256) fp32 tile from global into LDS ----
    if (threadIdx.x < 32) {   // wave 0 issues the descriptor
        const uint32_t lds_addr = (uint32_t)(uintptr_t)&tile[0]; // low 32b of LDS aperture addr

        v4u g0;
        g0[0] = 1u;                                           // count=1, user descriptor
        g0[1] = lds_addr;                                     // lds_addr (bytes)
        g0[2] = (uint32_t)ga;                                 // global_addr[31:0]
        g0[3] = (uint32_t)((ga >> 32) & 0x01FFFFFFu)          // global_addr[56:32]
              | (2u << 30);                                   // type=2 ("image")

        v8i g1;
        g1[0] = (2 << 16);                                    // data_size=4B, wg_mask=0
        g1[1] = (int)((IMG_W & 0xFFFF) << 16);                // tensor_dim0[15:0]
        g1[2] = (int)(((unsigned)IMG_W >> 16) | ((unsigned)(nrows & 0xFFFF) << 16)); // dim0 hi | dim1 lo
        g1[3] = (int)(((unsigned)nrows >> 16) | ((unsigned)IMG_W << 16));            // dim1 hi | tile_dim0
        g1[4] = nrows;                                        // tile_dim1 ; tile_dim2=0 (2D)
        g1[5] = IMG_W;                                        // tensor_dim0_stride[31:0]
        g1[6] = 0;                                            // stride0 hi | stride1 lo
        g1[7] = 0;                                            // stride1 hi

        v4i gz4 = {0, 0, 0, 0};                               // groups 2/3 unused (2D tile)
        v8i gz8 = {0, 0, 0, 0, 0, 0, 0, 0};                   // 6-arg form: extra group, zero-filled

        __builtin_amdgcn_tensor_load_to_lds(g0, g1, gz4, gz4, gz8, /*cpol=*/0);
        __builtin_amdgcn_s_wait_tensorcnt(0);
    }
    __syncthreads();

    // ---- compute: one wave per output row, 8 pixels/thread strided by 32 ----
    const int lane = threadIdx.x & 31;
    const int r    = threadIdx.x >> 5;           // 0..15
    const int y    = y0 + r;

    const int gym = (y == 0) ? 0 : (y - 1);
    const int gyp = (y == IMG_H - 1) ? (IMG_H - 1) : (y + 1);
    const int rm  = (gym - ystart) * IMG_W;
    const int rc  = (y   - ystart) * IMG_W;
    const int rp  = (gyp - ystart) * IMG_W;

    float* o = out + (size_t)plane * IMG_H * IMG_W + (size_t)y * IMG_W;

#pragma unroll
    for (int j = 0; j < 8; ++j) {
        const int xc = lane + 32 * j;
        const int xm = (xc == 0) ? 0 : (xc - 1);
        const int xp = (xc == IMG_W - 1) ? (IMG_W - 1) : (xc + 1);

        const float c = tile[rc + xc];
        float m = 0.0f;                              // zeroed center participates in max
        m = fmaxf(m, tile[rm + xm]);
        m = fmaxf(m, tile[rm + xc]);
        m = fmaxf(m, tile[rm + xp]);
        m = fmaxf(m, tile[rc + xm]);
        m = fmaxf(m, tile[rc + xp]);
        m = fmaxf(m, tile[rp + xm]);
        m = fmaxf(m, tile[rp + xc]);
        m = fmaxf(m, tile[rp + xp]);

        o[xc] = (c > m) ? c : 0.0f;
    }
}

extern "C" void kernel_launch(void* const* d_in, const int* in_sizes, int n_in,
                              void* d_out, int out_size, void* d_ws, size_t ws_size,
                              hipStream_t stream) {
    const float* x   = (const float*)d_in[0];
    float*       out = (float*)d_out;

    const int planes = in_sizes[0] / (IMG_H * IMG_W);   // 8*64 = 512
    const int tiles  = IMG_H / ROWS;                    // 16
    dim3 grid(planes * tiles);                          // 8192 blocks
    dim3 block(NTHREADS);                               // 512 threads = 16 waves

    nms3x3_kernel<<<grid, block, 0, stream>>>(x, out);
}